// LocalGrouper_54047868453479
// MI455X (gfx1250) — compile-verified
//
#include <hip/hip_runtime.h>
#include <hip/hip_bf16.h>

// ---------------------------------------------------------------------------
// PointNet LocalGrouper + PreExtraction for MI455X (gfx1250, wave32, WMMA).
// Pipeline: FPS -> kNN -> anchor-normalize (global std) -> GEMM(131->64) ->
// BN+ReLU -> GEMM(64->64) -> BN+ReLU -> GEMM(64->64) -> residual+ReLU ->
// max over k.  GEMMs use V_WMMA_F32_16X16X4_F32 (fully unrolled K loop) with
// the activation panel staged into LDS via GLOBAL_LOAD_ASYNC_TO_LDS_B128.
// ---------------------------------------------------------------------------

typedef __attribute__((ext_vector_type(2))) float v2f;
typedef __attribute__((ext_vector_type(8))) float v8f;

#define EPS_F 1e-5f

constexpr int B_  = 8;
constexpr int N_  = 4096;
constexpr int D_  = 64;
constexpr int S_  = 1024;   // groups
constexpr int K_  = 32;     // neighbors
constexpr int CIN  = 2 * D_ + 3;  // 131
constexpr int CINP = 132;         // padded to multiple of 4 (WMMA K step)
constexpr int MK  = B_ * S_ * K_; // 262144 GEMM columns
constexpr int AN  = D_ + 3;       // 67 anchor channels

// ---------------------------------------------------------------------------
// 1) Farthest point sampling: one block per batch, 1024 threads.
// ---------------------------------------------------------------------------
__global__ __launch_bounds__(1024) void fps_kernel(
    const float* __restrict__ xyz,      // [B,3,N]
    int* __restrict__ fps_idx,          // [B,S]
    float* __restrict__ new_xyz,        // [B,S,3]
    float* __restrict__ out0)           // [B,3,S]
{
  const int b   = blockIdx.x;
  const int tid = threadIdx.x;
  __shared__ float dist[N_];
  __shared__ float rv[1024];
  __shared__ int   ri[1024];
  __shared__ int   s_far;

  const float* xb = xyz + (size_t)b * 3 * N_;
  for (int i = tid; i < N_; i += 1024) dist[i] = 1e10f;
  if (tid == 0) s_far = 0;
  __syncthreads();

  for (int s = 0; s < S_; ++s) {
    const int far = s_far;
    const float cx = xb[far], cy = xb[N_ + far], cz = xb[2 * N_ + far];
    if (tid == 0) {
      fps_idx[b * S_ + s] = far;
      new_xyz[((size_t)b * S_ + s) * 3 + 0] = cx;
      new_xyz[((size_t)b * S_ + s) * 3 + 1] = cy;
      new_xyz[((size_t)b * S_ + s) * 3 + 2] = cz;
      out0[(size_t)b * 3 * S_ + 0 * S_ + s] = cx;
      out0[(size_t)b * 3 * S_ + 1 * S_ + s] = cy;
      out0[(size_t)b * 3 * S_ + 2 * S_ + s] = cz;
    }
    float bv = -1e30f; int bi = 0x7fffffff;
    for (int i = tid; i < N_; i += 1024) {
      const float dx = xb[i] - cx, dy = xb[N_ + i] - cy, dz = xb[2 * N_ + i] - cz;
      const float d  = dx * dx + dy * dy + dz * dz;
      const float nd = fminf(dist[i], d);
      dist[i] = nd;
      if (nd > bv || (nd == bv && i < bi)) { bv = nd; bi = i; }
    }
    rv[tid] = bv; ri[tid] = bi;
    __syncthreads();
    for (int off = 512; off > 0; off >>= 1) {
      if (tid < off) {
        const float v2 = rv[tid + off]; const int i2 = ri[tid + off];
        if (v2 > rv[tid] || (v2 == rv[tid] && i2 < ri[tid])) { rv[tid] = v2; ri[tid] = i2; }
      }
      __syncthreads();
    }
    if (tid == 0) s_far = ri[0];
    __syncthreads();
  }
}

// ---------------------------------------------------------------------------
// 2) kNN: one block per (b,s) centroid; distances in LDS, 32 masked argmins.
// ---------------------------------------------------------------------------
__global__ __launch_bounds__(256) void knn_kernel(
    const float* __restrict__ xyz,        // [B,3,N]
    const float* __restrict__ new_xyz,    // [B,S,3]
    int* __restrict__ knn_idx)            // [B,S,K]
{
  const int bs  = blockIdx.x;     // b*S + s
  const int b   = bs / S_;
  const int tid = threadIdx.x;
  __shared__ float d[N_];
  __shared__ float rv[256];
  __shared__ int   ri[256];

  const float* xb = xyz + (size_t)b * 3 * N_;
  const float cx = new_xyz[(size_t)bs * 3 + 0];
  const float cy = new_xyz[(size_t)bs * 3 + 1];
  const float cz = new_xyz[(size_t)bs * 3 + 2];

  for (int i = tid; i < N_; i += 256) {
    const float dx = xb[i] - cx, dy = xb[N_ + i] - cy, dz = xb[2 * N_ + i] - cz;
    d[i] = dx * dx + dy * dy + dz * dz;
  }
  __syncthreads();

  for (int j = 0; j < K_; ++j) {
    float bv = 1e30f; int bi = 0x7fffffff;
    for (int i = tid; i < N_; i += 256) {
      const float v = d[i];
      if (v < bv || (v == bv && i < bi)) { bv = v; bi = i; }
    }
    rv[tid] = bv; ri[tid] = bi;
    __syncthreads();
    for (int off = 128; off > 0; off >>= 1) {
      if (tid < off) {
        const float v2 = rv[tid + off]; const int i2 = ri[tid + off];
        if (v2 < rv[tid] || (v2 == rv[tid] && i2 < ri[tid])) { rv[tid] = v2; ri[tid] = i2; }
      }
      __syncthreads();
    }
    if (tid == 0) { knn_idx[(size_t)bs * K_ + j] = ri[0]; d[ri[0]] = 1e30f; }
    __syncthreads();
  }
}

// ---------------------------------------------------------------------------
// 3) Per-batch sum / sumsq of diff (for std with ddof=1).
// ---------------------------------------------------------------------------
__global__ __launch_bounds__(256) void group_stats_kernel(
    const float* __restrict__ xyz, const float* __restrict__ feature,
    const int* __restrict__ fps_idx, const int* __restrict__ knn_idx,
    float* __restrict__ bstats)   // [B,2] zero-initialized
{
  const int bs  = blockIdx.x;
  const int b   = bs / S_;
  const int tid = threadIdx.x;
  const float* fb = feature + (size_t)b * D_ * N_;
  const float* xb = xyz + (size_t)b * 3 * N_;
  const int fi = fps_idx[bs];

  __shared__ float anchor[AN];
  for (int c = tid; c < AN; c += 256)
    anchor[c] = (c < D_) ? fb[(size_t)c * N_ + fi] : xb[(size_t)(c - D_) * N_ + fi];
  __syncthreads();

  float s = 0.f, ss = 0.f;
  for (int t = tid; t < K_ * AN; t += 256) {
    const int j = t / AN, c = t % AN;
    const int gi = knn_idx[(size_t)bs * K_ + j];
    const float gv = (c < D_) ? fb[(size_t)c * N_ + gi] : xb[(size_t)(c - D_) * N_ + gi];
    const float df = gv - anchor[c];
    s += df; ss += df * df;
  }
  __shared__ float r0[256], r1[256];
  r0[tid] = s; r1[tid] = ss;
  __syncthreads();
  for (int off = 128; off > 0; off >>= 1) {
    if (tid < off) { r0[tid] += r0[tid + off]; r1[tid] += r1[tid + off]; }
    __syncthreads();
  }
  if (tid == 0) {
    atomicAdd(&bstats[b * 2 + 0], r0[0]);
    atomicAdd(&bstats[b * 2 + 1], r1[0]);
  }
}

// ---------------------------------------------------------------------------
// 4) Build GEMM input Xin [CINP][MK]: rows 0..66 affine-normalized diff,
//    rows 67..130 anchor feature, row 131 zero pad.
// ---------------------------------------------------------------------------
__global__ __launch_bounds__(256) void build_x_kernel(
    const float* __restrict__ xyz, const float* __restrict__ feature,
    const int* __restrict__ fps_idx, const int* __restrict__ knn_idx,
    const float* __restrict__ bstats,
    const float* __restrict__ alpha, const float* __restrict__ beta,
    float* __restrict__ Xin)
{
  const int bs  = blockIdx.x;
  const int b   = bs / S_;
  const int tid = threadIdx.x;
  const float* fb = feature + (size_t)b * D_ * N_;
  const float* xb = xyz + (size_t)b * 3 * N_;
  const int fi = fps_idx[bs];

  __shared__ float anchor[AN];
  __shared__ float inv_std;
  for (int c = tid; c < AN; c += 256)
    anchor[c] = (c < D_) ? fb[(size_t)c * N_ + fi] : xb[(size_t)(c - D_) * N_ + fi];
  if (tid == 0) {
    const float cnt = (float)S_ * K_ * AN;
    const float sum = bstats[b * 2 + 0], sq = bstats[b * 2 + 1];
    const float var = (sq - sum * sum / cnt) / (cnt - 1.0f);
    inv_std = 1.0f / (sqrtf(fmaxf(var, 0.f)) + EPS_F);
  }
  __syncthreads();

  const size_t colbase = (size_t)bs * K_;
  for (int t = tid; t < K_ * CINP; t += 256) {
    const int j = t % K_, row = t / K_;
    float val;
    if (row < AN) {
      const int gi = knn_idx[(size_t)bs * K_ + j];
      const float gv = (row < D_) ? fb[(size_t)row * N_ + gi] : xb[(size_t)(row - D_) * N_ + gi];
      val = alpha[row] * ((gv - anchor[row]) * inv_std) + beta[row];
    } else if (row < CIN) {
      val = anchor[row - AN];     // anchor feature channels
    } else {
      val = 0.f;                  // K padding row
    }
    Xin[(size_t)row * MK + colbase + j] = val;
  }
}

// ---------------------------------------------------------------------------
// 5) GEMM: Y[64][MK] = W[64][Cin] @ X[CinPad][MK] + bias  (f32 WMMA 16x16x4).
//    Block = 256 threads (8 waves). Block tile = 64 rows x 32 cols; each wave
//    owns one 16x16 output tile. The X panel is DMA'd straight into LDS with
//    GLOBAL_LOAD_ASYNC_TO_LDS_B128 (GVS mode: SGPR64 base + per-lane byte
//    offset), completion enforced with s_wait_asynccnt before the barrier.
//    The W panel (needs zero padding) uses guarded loads.  Cin/CinPad are
//    compile-time so the K loop fully unrolls (33 resp. 16 WMMAs, no
//    remainder loop, software-pipelined ds loads).
//    A layout (16x4 f32): lane l -> M = l&15, VGPR j -> K = kk + 2*(l>>4) + j.
//    B layout (4x16 f32): lane l -> N = l&15, VGPR j -> K = kk + 2*(l>>4) + j.
//    C/D (16x16 f32):     lane l, VGPR v -> M = v + 8*(l>>4), N = l&15.
// ---------------------------------------------------------------------------
template <int CinT, int CinPadT>
__global__ __launch_bounds__(256) void gemm_kernel(
    const float* __restrict__ X, const float* __restrict__ W,
    const float* __restrict__ bias, float* __restrict__ Y)
{
  __shared__ alignas(16) float s_w[64 * CinPadT];
  __shared__ alignas(16) float s_x[CinPadT * 32];
  const int tid = threadIdx.x;
  const size_t colbase = (size_t)blockIdx.x * 32;
  const float* xcol = X + colbase;   // uniform -> SGPR pair

  // Async-stage the X panel: CinPad rows x 32 cols as 16-byte transfers.
  // Row stride in memory is MK floats; both sides are 16B aligned.
  constexpr int NVEC = CinPadT * 8;  // 8 float4 per 32-col row
  for (int i = tid; i < NVEC; i += 256) {
    const int r = i >> 3, c4 = i & 7;
    const unsigned voff = (unsigned)(((size_t)r * MK + (size_t)c4 * 4) * sizeof(float));
    const unsigned lds  = (unsigned)(uintptr_t)&s_x[r * 32 + c4 * 4];
    asm volatile("global_load_async_to_lds_b128 %0, %1, %2"
                 :: "v"(lds), "v"(voff), "s"(xcol)
                 : "memory");
  }
  // W panel: guarded loads (zero-fill the K padding column).
  for (int i = tid; i < 64 * CinPadT; i += 256) {
    const int r = i / CinPadT, c = i % CinPadT;
    s_w[i] = (c < CinT) ? W[(size_t)r * CinT + c] : 0.f;
  }
  asm volatile("s_wait_asynccnt 0x0" ::: "memory");
  __syncthreads();

  const int lane = tid & 31;
  const int wave = tid >> 5;
  const int m0 = (wave & 3) * 16;     // output row tile
  const int n0 = (wave >> 2) * 16;    // output col tile
  const int mr = lane & 15;
  const int kh = lane >> 4;

  v8f acc = {};
#pragma unroll
  for (int kk = 0; kk < CinPadT; kk += 4) {
    v2f a, bb;
    a.x  = s_w[(m0 + mr) * CinPadT + kk + 2 * kh + 0];
    a.y  = s_w[(m0 + mr) * CinPadT + kk + 2 * kh + 1];
    bb.x = s_x[(kk + 2 * kh + 0) * 32 + n0 + mr];
    bb.y = s_x[(kk + 2 * kh + 1) * 32 + n0 + mr];
    acc = __builtin_amdgcn_wmma_f32_16x16x4_f32(
        /*neg_a=*/false, a, /*neg_b=*/false, bb,
        /*c_mod=*/(short)0, acc, /*reuse_a=*/false, /*reuse_b=*/false);
  }

#pragma unroll
  for (int v = 0; v < 8; ++v) {
    const int m = m0 + v + 8 * kh;
    Y[(size_t)m * MK + colbase + n0 + mr] = acc[v] + bias[m];
  }
}

// ---------------------------------------------------------------------------
// 6) BatchNorm (training batch stats, biased var) stats + apply(+ReLU).
// ---------------------------------------------------------------------------
__global__ __launch_bounds__(256) void bn_stats_kernel(
    const float* __restrict__ Y, float* __restrict__ st)  // st[64][2]
{
  const int c = blockIdx.x, tid = threadIdx.x;
  const float* p = Y + (size_t)c * MK;
  float s = 0.f, ss = 0.f;
  for (int i = tid; i < MK; i += 256) { const float v = p[i]; s += v; ss += v * v; }
  __shared__ float r0[256], r1[256];
  r0[tid] = s; r1[tid] = ss;
  __syncthreads();
  for (int off = 128; off > 0; off >>= 1) {
    if (tid < off) { r0[tid] += r0[tid + off]; r1[tid] += r1[tid + off]; }
    __syncthreads();
  }
  if (tid == 0) {
    const float mu = r0[0] / (float)MK;
    st[c * 2 + 0] = mu;
    st[c * 2 + 1] = r1[0] / (float)MK - mu * mu;
  }
}

__global__ __launch_bounds__(256) void bn_relu_kernel(
    const float* __restrict__ Y, const float* __restrict__ st,
    const float* __restrict__ g, const float* __restrict__ be,
    float* __restrict__ Yn)
{
  const size_t total = (size_t)64 * MK;
  for (size_t i = (size_t)blockIdx.x * blockDim.x + threadIdx.x; i < total;
       i += (size_t)gridDim.x * blockDim.x) {
    const int c = (int)(i / MK);
    const float v = g[c] * (Y[i] - st[c * 2]) * rsqrtf(st[c * 2 + 1] + EPS_F) + be[c];
    Yn[i] = fmaxf(v, 0.f);
  }
}

// ---------------------------------------------------------------------------
// 7) Residual + ReLU + max over k, write [B,64,S].
// ---------------------------------------------------------------------------
__global__ __launch_bounds__(256) void residual_max_kernel(
    const float* __restrict__ Y2, const float* __restrict__ X0n,
    float* __restrict__ out2)
{
  const size_t total = (size_t)B_ * 64 * S_;
  const size_t t = (size_t)blockIdx.x * blockDim.x + threadIdx.x;
  if (t >= total) return;
  const int s = (int)(t % S_);
  const int c = (int)((t / S_) % 64);
  const int b = (int)(t / ((size_t)64 * S_));
  const size_t col = ((size_t)b * S_ + s) * K_;
  const float* y = Y2  + (size_t)c * MK + col;
  const float* x = X0n + (size_t)c * MK + col;
  float m = 0.f;   // relu output is >= 0
  for (int j = 0; j < K_; ++j) m = fmaxf(m, fmaxf(y[j] + x[j], 0.f));
  out2[t] = m;
}

// ---------------------------------------------------------------------------
extern "C" void kernel_launch(void* const* d_in, const int* in_sizes, int n_in,
                              void* d_out, int out_size, void* d_ws, size_t ws_size,
                              hipStream_t stream) {
  (void)in_sizes; (void)n_in; (void)out_size;
  const float* xyz     = (const float*)d_in[0];
  const float* feature = (const float*)d_in[1];
  const float* alpha   = (const float*)d_in[2];
  const float* beta    = (const float*)d_in[3];
  const float* w_tr    = (const float*)d_in[4];
  const float* b_tr    = (const float*)d_in[5];
  const float* g_tr    = (const float*)d_in[6];
  const float* be_tr   = (const float*)d_in[7];
  const float* w1      = (const float*)d_in[8];
  const float* b1      = (const float*)d_in[9];
  const float* g1      = (const float*)d_in[10];
  const float* be1     = (const float*)d_in[11];
  const float* w2      = (const float*)d_in[12];
  const float* b2      = (const float*)d_in[13];
  // d_in[14]=groups(1024), d_in[15]=k(32): baked in as S_, K_.

  float* out0 = (float*)d_out;                       // [B,3,S]
  float* out2 = (float*)d_out + (size_t)B_ * 3 * S_; // [B,64,S]

  // Workspace carve-out (assumes ws_size >= ~480 MB).
  char* w = (char*)d_ws;
  auto carve = [&](size_t bytes) -> void* {
    void* p = (void*)w;
    w += (bytes + 255) & ~(size_t)255;
    return p;
  };
  int*   fps_idx = (int*)  carve((size_t)B_ * S_ * sizeof(int));
  int*   knn_idx = (int*)  carve((size_t)B_ * S_ * K_ * sizeof(int));
  float* new_xyz = (float*)carve((size_t)B_ * S_ * 3 * sizeof(float));
  float* bstats  = (float*)carve((size_t)B_ * 2 * sizeof(float));
  float* bnst    = (float*)carve((size_t)64 * 2 * sizeof(float));
  float* Xin     = (float*)carve((size_t)CINP * MK * sizeof(float));
  float* X0      = (float*)carve((size_t)64 * MK * sizeof(float));
  float* X0n     = (float*)carve((size_t)64 * MK * sizeof(float));
  float* Y1      = (float*)carve((size_t)64 * MK * sizeof(float));
  float* Y1n     = (float*)carve((size_t)64 * MK * sizeof(float));
  float* Y2      = (float*)carve((size_t)64 * MK * sizeof(float));
  (void)ws_size;

  // 1) FPS (writes out0 + new_xyz + fps_idx)
  fps_kernel<<<B_, 1024, 0, stream>>>(xyz, fps_idx, new_xyz, out0);
  // 2) kNN
  knn_kernel<<<B_ * S_, 256, 0, stream>>>(xyz, new_xyz, knn_idx);
  // 3) per-batch std accumulators
  hipMemsetAsync(bstats, 0, (size_t)B_ * 2 * sizeof(float), stream);
  group_stats_kernel<<<B_ * S_, 256, 0, stream>>>(xyz, feature, fps_idx, knn_idx, bstats);
  // 4) build GEMM input
  build_x_kernel<<<B_ * S_, 256, 0, stream>>>(xyz, feature, fps_idx, knn_idx,
                                              bstats, alpha, beta, Xin);
  // 5) conv_tr (131 -> 64) + BN + ReLU
  gemm_kernel<CIN, CINP><<<MK / 32, 256, 0, stream>>>(Xin, w_tr, b_tr, X0);
  bn_stats_kernel<<<64, 256, 0, stream>>>(X0, bnst);
  bn_relu_kernel<<<4096, 256, 0, stream>>>(X0, bnst, g_tr, be_tr, X0n);
  // 6) conv1 (64 -> 64) + BN + ReLU
  gemm_kernel<64, 64><<<MK / 32, 256, 0, stream>>>(X0n, w1, b1, Y1);
  bn_stats_kernel<<<64, 256, 0, stream>>>(Y1, bnst);
  bn_relu_kernel<<<4096, 256, 0, stream>>>(Y1, bnst, g1, be1, Y1n);
  // 7) conv2 (64 -> 64), residual + ReLU + max over k
  gemm_kernel<64, 64><<<MK / 32, 256, 0, stream>>>(Y1n, w2, b2, Y2);
  residual_max_kernel<<<(B_ * 64 * S_ + 255) / 256, 256, 0, stream>>>(Y2, X0n, out2);
}